// FeatureMatching_26723286516108
// MI455X (gfx1250) — compile-verified
//
#include <hip/hip_runtime.h>
#include <hip/hip_bf16.h>

typedef __attribute__((ext_vector_type(2))) float v2f;
typedef __attribute__((ext_vector_type(8))) float v8f;

#define HW1 (192*192)
#define NPIX 9216     // 96*96
#define NF   144      // 16 channels * 9 taps
#define KSLICES 8     // key-dimension split of the similarity GEMM

// All feature maps / patch matrices use a pair-interleaved layout:
//   T[img][c/2][pix][2]  (flat float index = img_off + ((c>>1)*HW + pix)*2 + (c&1))
// so a WMMA fragment (features f, f+1 for one pixel) is a single b64 load.

// ---------------------------------------------------------------------------
// VGG normalize + conv3x3 (3->64) + bias + relu, zero 'SAME' pad. Direct FMA.
// ---------------------------------------------------------------------------
__global__ void conv1_kernel(const float* __restrict__ qin, const float* __restrict__ kin,
                             const float* __restrict__ w0, const float* __restrict__ b0,
                             float* __restrict__ out) {
    const float mean[3] = {0.485f, 0.456f, 0.406f};
    const float stdv[3] = {0.229f, 0.224f, 0.225f};
    int tid = blockIdx.x * blockDim.x + threadIdx.x;   // 2*36864*16 threads exactly
    int cg  = tid & 15;
    int pix = (tid >> 4) % HW1;
    int img = tid / (16 * HW1);
    const float* in = img ? kin : qin;
    int y = pix / 192, x = pix % 192;
    int obase = cg * 4;
    float a0 = 0.f, a1 = 0.f, a2 = 0.f, a3 = 0.f;
    for (int t = 0; t < 9; ++t) {
        int dy = t / 3 - 1, dx = t % 3 - 1;
        int yy = y + dy, xx = x + dx;
        bool valid = (yy >= 0) && (yy < 192) && (xx >= 0) && (xx < 192);
        int yyc = yy < 0 ? 0 : (yy > 191 ? 191 : yy);
        int xxc = xx < 0 ? 0 : (xx > 191 ? 191 : xx);
        for (int c = 0; c < 3; ++c) {
            float v = (in[c * HW1 + yyc * 192 + xxc] - mean[c]) / stdv[c];
            v = valid ? v : 0.f;
            a0 += w0[(obase + 0) * 27 + c * 9 + t] * v;
            a1 += w0[(obase + 1) * 27 + c * 9 + t] * v;
            a2 += w0[(obase + 2) * 27 + c * 9 + t] * v;
            a3 += w0[(obase + 3) * 27 + c * 9 + t] * v;
        }
    }
    a0 += b0[obase + 0]; a1 += b0[obase + 1]; a2 += b0[obase + 2]; a3 += b0[obase + 3];
    float* o = out + (size_t)img * 64 * HW1;
    size_t p0 = ((size_t)(obase >> 1) * HW1 + pix) * 2;
    size_t p1 = ((size_t)((obase >> 1) + 1) * HW1 + pix) * 2;
    o[p0 + 0] = a0 > 0.f ? a0 : 0.f;
    o[p0 + 1] = a1 > 0.f ? a1 : 0.f;
    o[p1 + 0] = a2 > 0.f ? a2 : 0.f;
    o[p1 + 1] = a3 > 0.f ? a3 : 0.f;
}

// ---------------------------------------------------------------------------
// Repack conv weights OIHW -> [o][tap][cin] so B fragments are b64 loads.
// ---------------------------------------------------------------------------
__global__ void repack_w_kernel(const float* __restrict__ w, float* __restrict__ wpk,
                                int Cin, int n) {
    int tid = blockIdx.x * blockDim.x + threadIdx.x;   // n = Cout*9*Cin
    if (tid >= n) return;
    int c = tid % Cin;
    int t = (tid / Cin) % 9;
    int o = tid / (Cin * 9);
    wpk[tid] = w[((size_t)o * Cin + c) * 9 + t];
}

// ---------------------------------------------------------------------------
// conv3x3 as implicit GEMM on V_WMMA_F32_16X16X4_F32.
// Wave tile: M = 16 row-contiguous pixels, N = 16 out channels, K = Cin*9.
// Fragments staged in groups of 4 K-chunks (8 b64 loads in flight per wait).
// Boundary pad via clamped address + cndmask select (no EXEC manipulation).
// ---------------------------------------------------------------------------
__global__ void conv3x3_wmma(const float* __restrict__ in, float* __restrict__ out,
                             const float* __restrict__ wpk, const float* __restrict__ b,
                             int Cin, int Cout, int H, int W, int relu) {
    int lane = threadIdx.x & 31;
    int wid  = threadIdx.x >> 5;
    int PT = (H * W) >> 4;
    int NG = Cout >> 4;
    int wt  = blockIdx.x * 4 + wid;      // exact grids, no guard
    int img = wt / (PT * NG);
    int rem = wt % (PT * NG);
    int ptile = rem / NG;
    int ng    = rem % NG;
    const v2f* inV  = (const v2f*)(in + (size_t)img * Cin * H * W);
    float*     outI = out + (size_t)img * Cout * H * W;
    int pix0 = ptile << 4;
    int y0 = pix0 / W;
    int x0 = pix0 % W;
    int hh = lane >> 4;
    int lm = lane & 15;
    int o  = ng * 16 + lm;               // N index / out channel
    const v2f* wV = (const v2f*)wpk;
    const v2f zero2 = {0.f, 0.f};

    v8f acc = {0.f, 0.f, 0.f, 0.f, 0.f, 0.f, 0.f, 0.f};
    for (int t = 0; t < 9; ++t) {
        int dy = t / 3 - 1, dx = t % 3 - 1;
        int yy = y0 + dy;
        int xx = x0 + lm + dx;
        bool valid = (yy >= 0) && (yy < H) && (xx >= 0) && (xx < W);
        int yyc = yy < 0 ? 0 : (yy > H - 1 ? H - 1 : yy);
        int xxc = xx < 0 ? 0 : (xx > W - 1 ? W - 1 : xx);
        const v2f* ap = inV + hh * (size_t)(H * W) + yyc * W + xxc;
        const v2f* bp = wV + ((size_t)o * 9 + t) * (Cin >> 1) + hh;
        for (int g = 0; g < (Cin >> 4); ++g) {         // group = 4 K-chunks = 16 chans
            v2f a[4], bb[4];
#pragma unroll
            for (int u = 0; u < 4; ++u) {
                int c2 = g * 4 + u;
                a[u]  = ap[(size_t)c2 * 2 * H * W];
                bb[u] = bp[(size_t)c2 * 2];
            }
#pragma unroll
            for (int u = 0; u < 4; ++u) {
                v2f av = valid ? a[u] : zero2;
                acc = __builtin_amdgcn_wmma_f32_16x16x4_f32(
                          false, av, false, bb[u], (short)0, acc, false, false);
            }
        }
    }
    float bias = b[o];
    size_t obase = ((size_t)(o >> 1) * H * W) * 2 + (o & 1);
#pragma unroll
    for (int r = 0; r < 8; ++r) {
        int m = r + 8 * hh;
        float v = acc[r] + bias;
        if (relu) v = v > 0.f ? v : 0.f;
        outI[obase + (size_t)(pix0 + m) * 2] = v;
    }
}

// ---------------------------------------------------------------------------
// 2x2/2 maxpool in paired layout.
// ---------------------------------------------------------------------------
__global__ void maxpool_kernel(const float* __restrict__ in, float* __restrict__ out) {
    int tid = blockIdx.x * blockDim.x + threadIdx.x;  // 2*64*96*96 exactly
    int s = tid & 1;
    int rest = tid >> 1;
    int x = rest % 96;
    int y = (rest / 96) % 96;
    int cp = rest / (96 * 96);
    const float* p = in + ((size_t)cp * HW1 + (2 * y) * 192 + 2 * x) * 2 + s;
    out[tid] = fmaxf(fmaxf(p[0], p[2]), fmaxf(p[384], p[386]));
}

// ---------------------------------------------------------------------------
// 3x3 unfold (reflect pad) + L2 normalize over f=144 -> paired [img][72][NPIX][2].
// ---------------------------------------------------------------------------
__global__ void patch_norm_kernel(const float* __restrict__ fm, float* __restrict__ p) {
    int tid = blockIdx.x * blockDim.x + threadIdx.x;  // 2*9216 exactly
    int pix = tid % NPIX;
    int img = tid / NPIX;
    const float* f = fm + (size_t)img * 16 * NPIX;
    float* po = p + (size_t)img * NF * NPIX;
    int y = pix / 96, x = pix % 96;
    float ss = 0.f;
    for (int c = 0; c < 16; ++c)
        for (int t = 0; t < 9; ++t) {
            int yy = y + t / 3 - 1; yy = yy < 0 ? -yy : (yy > 95 ? 190 - yy : yy);
            int xx = x + t % 3 - 1; xx = xx < 0 ? -xx : (xx > 95 ? 190 - xx : xx);
            float v = f[((size_t)(c >> 1) * NPIX + yy * 96 + xx) * 2 + (c & 1)];
            ss += v * v;
        }
    float inv = 1.f / fmaxf(sqrtf(ss), 1e-12f);
    for (int c = 0; c < 16; ++c)
        for (int t = 0; t < 9; ++t) {
            int yy = y + t / 3 - 1; yy = yy < 0 ? -yy : (yy > 95 ? 190 - yy : yy);
            int xx = x + t % 3 - 1; xx = xx < 0 ? -xx : (xx > 95 ? 190 - xx : xx);
            float v = f[((size_t)(c >> 1) * NPIX + yy * 96 + xx) * 2 + (c & 1)];
            int fi = c * 9 + t;
            po[((size_t)(fi >> 1) * NPIX + pix) * 2 + (fi & 1)] = v * inv;
        }
}

// ---------------------------------------------------------------------------
// Similarity partial kernel: grid (144 qblocks x 8 key-slices), 4 waves/block.
// Wave owns 16 query columns & one slice of 72 key tiles: query tile (144x16)
// in 72 VGPRs, key tiles streamed 2 at a time into two independent WMMA
// accumulator chains, fragments staged in groups of 6 (12 b64 loads / wait).
// Emits per-slice (max, argmax) partials; first-occurrence order preserved.
// ---------------------------------------------------------------------------
__global__ void similarity_kernel(const float* __restrict__ qn, const float* __restrict__ kn,
                                  float* __restrict__ pbest, int* __restrict__ pidx) {
    int lane = threadIdx.x & 31;
    int wid  = threadIdx.x >> 5;
    int q0 = (blockIdx.x * 4 + wid) * 16;   // 144*4*16 = 9216 queries
    int slice = blockIdx.y;                 // 0..7, keys [slice*1152, +1152)
    int hh = lane >> 4;
    int lm = lane & 15;
    const v2f* qV = (const v2f*)qn;         // [72][NPIX] pairs
    const v2f* kV = (const v2f*)kn;

    v2f breg[36];
#pragma unroll
    for (int j = 0; j < 36; ++j)
        breg[j] = qV[(size_t)(2 * j + hh) * NPIX + q0 + lm];

    float best = -3.4e38f;
    int bidx = 0;
    int kt0 = slice * (576 / KSLICES);
    for (int kt = kt0; kt < kt0 + 576 / KSLICES; kt += 2) {
        int kb0 = kt * 16, kb1 = kb0 + 16;
        v8f acc0 = {0.f, 0.f, 0.f, 0.f, 0.f, 0.f, 0.f, 0.f};
        v8f acc1 = {0.f, 0.f, 0.f, 0.f, 0.f, 0.f, 0.f, 0.f};
        for (int g = 0; g < 6; ++g) {       // 6 groups of 6 K-chunks
            v2f a0[6], a1[6];
#pragma unroll
            for (int u = 0; u < 6; ++u) {
                size_t row = (size_t)(2 * (g * 6 + u) + hh) * NPIX;
                a0[u] = kV[row + kb0 + lm];
                a1[u] = kV[row + kb1 + lm];
            }
#pragma unroll
            for (int u = 0; u < 6; ++u) {
                int j = g * 6 + u;
                acc0 = __builtin_amdgcn_wmma_f32_16x16x4_f32(
                           false, a0[u], false, breg[j], (short)0, acc0, false, false);
                acc1 = __builtin_amdgcn_wmma_f32_16x16x4_f32(
                           false, a1[u], false, breg[j], (short)0, acc1, false, false);
            }
        }
#pragma unroll
        for (int r = 0; r < 8; ++r) {       // keys ascend -> '>' keeps first max
            float v = acc0[r];
            int ki = kb0 + r + 8 * hh;
            if (v > best) { best = v; bidx = ki; }
        }
#pragma unroll
        for (int r = 0; r < 8; ++r) {
            float v = acc1[r];
            int ki = kb1 + r + 8 * hh;
            if (v > best) { best = v; bidx = ki; }
        }
    }
    // combine lane halves (same query column); tie -> smaller key index
    float ob = __shfl_xor(best, 16, 32);
    int   oi = __shfl_xor(bidx, 16, 32);
    if (ob > best || (ob == best && oi < bidx)) { best = ob; bidx = oi; }
    if (lane < 16) {
        pbest[(size_t)slice * NPIX + q0 + lane] = best;
        pidx [(size_t)slice * NPIX + q0 + lane] = bidx;
    }
}

// ---------------------------------------------------------------------------
// Combine the 8 per-slice partials. Slices partition keys in ascending order,
// so scanning s ascending with strict '>' keeps the first-occurrence argmax.
// ---------------------------------------------------------------------------
__global__ void argmax_reduce_kernel(const float* __restrict__ pbest, const int* __restrict__ pidx,
                                     float* __restrict__ rel, int* __restrict__ idx) {
    int q = blockIdx.x * blockDim.x + threadIdx.x;   // 9216 exactly
    float best = pbest[q];
    int bi = pidx[q];
    for (int s = 1; s < KSLICES; ++s) {
        float v = pbest[(size_t)s * NPIX + q];
        int i2 = pidx[(size_t)s * NPIX + q];
        if (v > best) { best = v; bi = i2; }
    }
    rel[q] = best;
    idx[q] = bi;
}

// ---------------------------------------------------------------------------
// Workspace (floats): A[4718592] | B[4718592] | P[2654208] | WPK[147456]
//                     | PB[73728] | PI[73728]   (~47.3 MiB total)
// d_out: 9216 f32 relevance then 9216 i32 hard_indices.
// ---------------------------------------------------------------------------
extern "C" void kernel_launch(void* const* d_in, const int* in_sizes, int n_in,
                              void* d_out, int out_size, void* d_ws, size_t ws_size,
                              hipStream_t stream) {
    const float* q  = (const float*)d_in[0];
    const float* k  = (const float*)d_in[1];
    const float* w0 = (const float*)d_in[2];
    const float* b0 = (const float*)d_in[3];
    const float* w2 = (const float*)d_in[4];
    const float* b2 = (const float*)d_in[5];
    const float* w5 = (const float*)d_in[6];
    const float* b5 = (const float*)d_in[7];
    const float* w7 = (const float*)d_in[8];
    const float* b7 = (const float*)d_in[9];
    const float* wm = (const float*)d_in[10];
    const float* bm = (const float*)d_in[11];

    float* ws  = (float*)d_ws;
    float* A   = ws;                  // 2*64*192*192 (paired layout)
    float* B   = ws + 4718592;        // ditto / reused at 96x96
    float* P   = ws + 9437184;        // 2*144*9216 normalized patches (paired)
    float* WPK = ws + 12091392;       // repacked weights, <=128*128*9
    float* PB  = ws + 12238848;       // 8*9216 partial maxima
    int*   PI  = (int*)(ws + 12312576); // 8*9216 partial argmax

    conv1_kernel     <<<4608, 256, 0, stream>>>(q, k, w0, b0, A);                    // A: f1
    repack_w_kernel  <<< 144, 256, 0, stream>>>(w2, WPK, 64, 64 * 9 * 64);
    conv3x3_wmma     <<<4608, 128, 0, stream>>>(A, B, WPK, b2, 64, 64, 192, 192, 1); // B: f2
    maxpool_kernel   <<<4608, 256, 0, stream>>>(B, A);                               // A: pooled
    repack_w_kernel  <<< 288, 256, 0, stream>>>(w5, WPK, 64, 128 * 9 * 64);
    conv3x3_wmma     <<<2304, 128, 0, stream>>>(A, B, WPK, b5, 64, 128, 96, 96, 1);  // B: f3
    repack_w_kernel  <<< 576, 256, 0, stream>>>(w7, WPK, 128, 128 * 9 * 128);
    conv3x3_wmma     <<<2304, 128, 0, stream>>>(B, A, WPK, b7, 128, 128, 96, 96, 1); // A: f4
    repack_w_kernel  <<<  72, 256, 0, stream>>>(wm, WPK, 128, 16 * 9 * 128);
    conv3x3_wmma     <<< 288, 128, 0, stream>>>(A, B, WPK, bm, 128, 16, 96, 96, 0);  // B: fm
    patch_norm_kernel<<<  72, 256, 0, stream>>>(B, P);                               // P: qn|kn
    similarity_kernel<<<dim3(144, KSLICES), 128, 0, stream>>>(P, P + (size_t)NF * NPIX, PB, PI);
    argmax_reduce_kernel<<<36, 256, 0, stream>>>(PB, PI, (float*)d_out, ((int*)d_out) + NPIX);
}